// RationalPartitionClassifier_15522011808346
// MI455X (gfx1250) — compile-verified
//
#include <hip/hip_runtime.h>
#include <math.h>
#include <stdint.h>

typedef __attribute__((ext_vector_type(2))) float v2f;
typedef __attribute__((ext_vector_type(8))) float v8f;
typedef __attribute__((ext_vector_type(4))) int   v4i;
typedef __attribute__((ext_vector_type(2))) int   v2i;

#define N_NODES   100000
#define N_EDGES   1600000
#define N_FEAT    7
#define HIDDEN    128
#define N_CLASSES 50
#define NC_PAD    64

#if defined(__AMDGCN__) && __has_builtin(__builtin_amdgcn_global_load_async_to_lds_b128)
#define HAS_ASYNC_LDS_B128 1
#endif
#if defined(__AMDGCN__) && __has_builtin(__builtin_amdgcn_global_load_async_to_lds_b64)
#define HAS_ASYNC_LDS_B64 1
#endif

__device__ __forceinline__ void wait_async_all() {
#if defined(__AMDGCN__)
#if __has_builtin(__builtin_amdgcn_s_wait_asynccnt)
    __builtin_amdgcn_s_wait_asynccnt(0);
#else
    asm volatile("s_wait_asynccnt 0x0" ::: "memory");
#endif
#endif
}

// ---------------- degree / normalization ----------------

__global__ void k_init_deg(float* __restrict__ deg, int n) {
    int i = blockIdx.x * blockDim.x + threadIdx.x;
    if (i < n) deg[i] = 1.0f;                    // self-loop contribution
}

__global__ void k_deg_scatter(const int* __restrict__ dst, float* __restrict__ deg, int ne) {
    int e = blockIdx.x * blockDim.x + threadIdx.x;
    if (e < ne) atomicAdd(&deg[dst[e]], 1.0f);
}

__global__ void k_rsqrt_inplace(float* __restrict__ deg, int n) {
    int i = blockIdx.x * blockDim.x + threadIdx.x;
    if (i < n) deg[i] = rsqrtf(deg[i]);          // deg >= 1 guaranteed
}

// ---------------- layer 1: hs1 = (x @ W1) * dis  (K=7, VALU) ----------------

__global__ void k_gemm1_scale(const float* __restrict__ x, const float* __restrict__ W1,
                              const float* __restrict__ dis, float* __restrict__ out, int n) {
    int idx = blockIdx.x * blockDim.x + threadIdx.x;
    if (idx >= n * HIDDEN) return;
    int node = idx >> 7;
    int col  = idx & (HIDDEN - 1);
    const float* xr = x + node * N_FEAT;
    float s = 0.0f;
#pragma unroll
    for (int k = 0; k < N_FEAT; ++k) s = fmaf(xr[k], W1[k * HIDDEN + col], s);
    out[idx] = s * dis[node];
}

// ---------------- edge scatter: agg[dst] += hs[src]  (one wave per edge) ----------------

__global__ void k_scatter_add(const int* __restrict__ src, const int* __restrict__ dst,
                              const float* __restrict__ hs, float* __restrict__ agg, int ne) {
    int idx = blockIdx.x * blockDim.x + threadIdx.x;   // ne*32 work items
    if (idx >= ne * 32) return;
    int e = idx >> 5;                                  // edge (uniform per wave)
    int q = idx & 31;                                  // 4-float chunk within 128 features
    int s = src[e], d = dst[e];
    const float4 v = *(const float4*)(hs + (size_t)s * HIDDEN + q * 4);
    float* p = agg + (size_t)d * HIDDEN + q * 4;
    atomicAdd(p + 0, v.x);
    atomicAdd(p + 1, v.y);
    atomicAdd(p + 2, v.z);
    atomicAdd(p + 3, v.w);
}

// ---------------- h = relu(agg * dis + b)  in place ----------------

__global__ void k_bias_relu(float* __restrict__ buf, const float* __restrict__ dis,
                            const float* __restrict__ b, int n) {
    int idx = blockIdx.x * blockDim.x + threadIdx.x;
    if (idx >= n * HIDDEN) return;
    int node = idx >> 7;
    int col  = idx & (HIDDEN - 1);
    float v = fmaf(buf[idx], dis[node], b[col]);
    buf[idx] = fmaxf(v, 0.0f);
}

// ---------------- layer 2 GEMM: hs2 = (Hin @ W[128x128]) * dis  (fp32 WMMA) ----------------
// One block = 8 waves; block covers 16 rows x 128 cols; wave w does cols [16w,16w+16).
// n must be a multiple of 16 (100000 = 6250*16).

__global__ void __launch_bounds__(256) k_gemm_wmma_h2h(const float* __restrict__ Hin,
                                                       const float* __restrict__ W,
                                                       const float* __restrict__ dis,
                                                       float* __restrict__ Hout, int n) {
    __shared__ float lw[HIDDEN * HIDDEN];              // 64 KB weight stage
    int tid = threadIdx.x;
#ifdef HAS_ASYNC_LDS_B128
    // async global->LDS DMA, 16B chunks, bypasses VGPRs (ASYNCcnt-tracked)
    for (int i = tid; i < (HIDDEN * HIDDEN) / 4; i += 256) {
        __builtin_amdgcn_global_load_async_to_lds_b128(
            (__attribute__((address_space(1))) v4i*)(W + i * 4),
            (__attribute__((address_space(3))) v4i*)(lw + i * 4),
            0, 0);
    }
    wait_async_all();
#else
    for (int i = tid; i < HIDDEN * HIDDEN; i += 256) lw[i] = W[i];
#endif
    __syncthreads();

    int wave = tid >> 5, lane = tid & 31;
    int lm = lane & 15, kh = lane >> 4;                // A frag: row lm, K pair 2*kh
    int rbase = blockIdx.x * 16;
    int cbase = wave * 16;
    int row = rbase + lm;

    const float* ap = Hin + (size_t)row * HIDDEN + 2 * kh;      // 8B-aligned pair loads
    const float* bp = lw + (2 * kh) * HIDDEN + cbase + lm;

    v8f c = {};
#pragma unroll 8
    for (int kk = 0; kk < HIDDEN; kk += 4) {
        v2f a = *(const v2f*)(ap + kk);                // {A[m][k], A[m][k+1]}
        v2f b;
        b.x = bp[kk * HIDDEN];                         // B[k][nc]
        b.y = bp[kk * HIDDEN + HIDDEN];                // B[k+1][nc]
        c = __builtin_amdgcn_wmma_f32_16x16x4_f32(false, a, false, b,
                                                  (short)0, c, false, false);
    }

    int colOut = cbase + lm;
    int r0 = rbase + (lane < 16 ? 0 : 8);
#pragma unroll
    for (int r = 0; r < 8; ++r) {
        int rowOut = r0 + r;
        Hout[(size_t)rowOut * HIDDEN + colOut] = c[r] * dis[rowOut];
    }
}

// ---------------- classifier GEMM: out = Hin @ Wc[128x50] + bc  (fp32 WMMA) ----------------
// Block = 8 waves as 2 row-groups x 4 col-groups -> 32 rows x 64 (padded) cols.
// n must be a multiple of 32 (100000 = 3125*32).

__global__ void __launch_bounds__(256) k_gemm_wmma_out(const float* __restrict__ Hin,
                                                       const float* __restrict__ Wc,
                                                       const float* __restrict__ bc,
                                                       float* __restrict__ out, int n) {
    __shared__ float lw[HIDDEN * NC_PAD];              // 32 KB, zero-padded cols 50..63
    int tid = threadIdx.x;
#ifdef HAS_ASYNC_LDS_B64
    // zero the pad columns with plain LDS stores (disjoint from async-written region)
    for (int i = tid; i < HIDDEN * NC_PAD; i += 256)
        if ((i & (NC_PAD - 1)) >= N_CLASSES) lw[i] = 0.0f;
    // async copy each 200B row as 25 x 8B chunks into the padded image
    for (int i = tid; i < HIDDEN * 25; i += 256) {
        int r = i / 25, c = i - r * 25;                // chunk c covers cols 2c..2c+1
        __builtin_amdgcn_global_load_async_to_lds_b64(
            (__attribute__((address_space(1))) v2i*)(Wc + r * N_CLASSES + c * 2),
            (__attribute__((address_space(3))) v2i*)(lw + r * NC_PAD + c * 2),
            0, 0);
    }
    wait_async_all();
#else
    for (int i = tid; i < HIDDEN * NC_PAD; i += 256) {
        int r = i >> 6, col = i & (NC_PAD - 1);
        lw[i] = (col < N_CLASSES) ? Wc[r * N_CLASSES + col] : 0.0f;
    }
#endif
    __syncthreads();

    int wave = tid >> 5, lane = tid & 31;
    int lm = lane & 15, kh = lane >> 4;
    int rw = wave >> 2, cw = wave & 3;
    int rbase = blockIdx.x * 32 + rw * 16;
    int cbase = cw * 16;
    int row = rbase + lm;

    const float* ap = Hin + (size_t)row * HIDDEN + 2 * kh;
    const float* bp = lw + (2 * kh) * NC_PAD + cbase + lm;

    v8f c = {};
#pragma unroll 8
    for (int kk = 0; kk < HIDDEN; kk += 4) {
        v2f a = *(const v2f*)(ap + kk);
        v2f b;
        b.x = bp[kk * NC_PAD];
        b.y = bp[kk * NC_PAD + NC_PAD];
        c = __builtin_amdgcn_wmma_f32_16x16x4_f32(false, a, false, b,
                                                  (short)0, c, false, false);
    }

    int colOut = cbase + lm;
    if (colOut < N_CLASSES) {                          // predicate stores only (WMMA already done)
        float bcv = bc[colOut];
        int r0 = rbase + (lane < 16 ? 0 : 8);
#pragma unroll
        for (int r = 0; r < 8; ++r)
            out[(size_t)(r0 + r) * N_CLASSES + colOut] = c[r] + bcv;
    }
}

// ---------------- row-wise log_softmax over 50 classes, in place (wave per row) ----------------

__global__ void k_logsoftmax(float* __restrict__ out, int n) {
    int wave = threadIdx.x >> 5, lane = threadIdx.x & 31;
    int row = blockIdx.x * 8 + wave;
    if (row >= n) return;
    float* p = out + (size_t)row * N_CLASSES;
    float a = p[lane];                                           // cols 0..31
    bool hi = (lane + 32) < N_CLASSES;
    float b = hi ? p[lane + 32] : -INFINITY;                     // cols 32..49
    float m = fmaxf(a, b);
    for (int off = 16; off; off >>= 1) m = fmaxf(m, __shfl_xor(m, off, 32));
    float s = expf(a - m) + (hi ? expf(b - m) : 0.0f);
    for (int off = 16; off; off >>= 1) s += __shfl_xor(s, off, 32);
    float lse = m + logf(s);
    p[lane] = a - lse;
    if (hi) p[lane + 32] = b - lse;
}

// ---------------- launch ----------------

extern "C" void kernel_launch(void* const* d_in, const int* in_sizes, int n_in,
                              void* d_out, int out_size, void* d_ws, size_t ws_size,
                              hipStream_t stream) {
    (void)in_sizes; (void)n_in; (void)out_size; (void)ws_size;

    const float* x  = (const float*)d_in[0];
    const int*   ei = (const int*)d_in[1];
    const float* W1 = (const float*)d_in[2];
    const float* b1 = (const float*)d_in[3];
    const float* W2 = (const float*)d_in[4];
    const float* b2 = (const float*)d_in[5];
    const float* Wc = (const float*)d_in[6];
    const float* bc = (const float*)d_in[7];
    float* out = (float*)d_out;

    const int n = N_NODES, ne = N_EDGES;
    const int* src = ei;            // edge_index[0]
    const int* dst = ei + ne;       // edge_index[1]

    char* ws = (char*)d_ws;
    float* dis = (float*)ws;                                     // n floats (deg -> rsqrt)
    size_t off = (((size_t)n * sizeof(float)) + 255) & ~(size_t)255;
    const size_t hbytes = (size_t)n * HIDDEN * sizeof(float);    // 51.2 MB
    float* bufA = (float*)(ws + off);                            // hs
    float* bufB = (float*)(ws + off + hbytes);                   // agg / h

    const int B = 256;
    const int gN  = (n + B - 1) / B;
    const int gE  = (ne + B - 1) / B;
    const int gNH = (n * HIDDEN + B - 1) / B;
    const int gSC = (ne * 32 + B - 1) / B;

    // normalization: dis = rsqrt(1 + indegree)
    k_init_deg<<<gN, B, 0, stream>>>(dis, n);
    k_deg_scatter<<<gE, B, 0, stream>>>(dst, dis, ne);
    k_rsqrt_inplace<<<gN, B, 0, stream>>>(dis, n);

    // ---- layer 1 ----
    k_gemm1_scale<<<gNH, B, 0, stream>>>(x, W1, dis, bufA, n);                  // hs1 = (x@W1)*dis
    (void)hipMemcpyAsync(bufB, bufA, hbytes, hipMemcpyDeviceToDevice, stream);  // agg = hs (self-loops)
    k_scatter_add<<<gSC, B, 0, stream>>>(src, dst, bufA, bufB, ne);             // agg[dst] += hs[src]
    k_bias_relu<<<gNH, B, 0, stream>>>(bufB, dis, b1, n);                       // h1 = relu(agg*dis + b1)

    // ---- layer 2 ----
    k_gemm_wmma_h2h<<<n / 16, B, 0, stream>>>(bufB, W2, dis, bufA, n);          // hs2 = (h1@W2)*dis
    (void)hipMemcpyAsync(bufB, bufA, hbytes, hipMemcpyDeviceToDevice, stream);  // agg = hs (self-loops)
    k_scatter_add<<<gSC, B, 0, stream>>>(src, dst, bufA, bufB, ne);
    k_bias_relu<<<gNH, B, 0, stream>>>(bufB, dis, b2, n);                       // h2

    // ---- classifier + log_softmax ----
    k_gemm_wmma_out<<<n / 32, B, 0, stream>>>(bufB, Wc, bc, out, n);            // logits -> d_out
    k_logsoftmax<<<(n + 7) / 8, B, 0, stream>>>(out, n);
}